// AStarScanStrategy_7662221656538
// MI455X (gfx1250) — compile-verified
//
#include <hip/hip_runtime.h>

// Problem constants (from the reference):
//   features: [B=16, C=64, H=192, W=192]  (f32)
//   path_idx: [B, P=128, L=512]           (int)
//   W_m: [C, C] f32, b_m: [C] f32
//   out: [B, C, H, W] f32
//
// Key identity: every path step landing on pixel p gathers the SAME feature
// row, so scatter(cnt copies of y_p)/clip(cnt,1) == y_p when cnt>0 else 0,
// where y_p = feat[:,p] @ W_m + b_m.  => masked per-pixel 64x64 linear map.

typedef __attribute__((ext_vector_type(2))) float v2f;
typedef __attribute__((ext_vector_type(8))) float v8f;

#define BATCHES 16
#define CCH     64
#define HWN     36864      // 192*192
#define PLN     65536      // 128*512 path steps per batch
#define TILES_PER_BATCH (HWN / 16)        // 2304 pixel tiles of 16
#define BLOCKS_PER_BATCH 8
#define WAVES_PER_BATCH (BLOCKS_PER_BATCH * 8)  // 64 -> 36 tiles per wave

// ---------------------------------------------------------------------------
// Kernel 1: zero the visit-flag buffer (B*HW bytes in d_ws), 16B per thread.
__global__ void astar_zero_flags(uint4* __restrict__ flags4, int n4) {
    int i = blockIdx.x * blockDim.x + threadIdx.x;
    if (i < n4) flags4[i] = make_uint4(0u, 0u, 0u, 0u);
}

// ---------------------------------------------------------------------------
// Kernel 2: mark visited pixels. Plain byte stores of the constant 1 are
// race-free (all writers store the same value) -> no atomics, deterministic.
__global__ void astar_set_flags(const int* __restrict__ path_idx,
                                unsigned char* __restrict__ flags) {
    int i = blockIdx.x * blockDim.x + threadIdx.x;   // [0, B*P*L), exact grid
    int b   = i >> 16;                               // i / PLN (PLN = 65536)
    int idx = path_idx[i];                           // flattened r*W + c, < HW
    flags[b * HWN + idx] = 1;
}

// ---------------------------------------------------------------------------
// Kernel 3: masked per-pixel linear map via V_WMMA_F32_16X16X4_F32.
//   D[M=chan_out(16), N=pixel(16)] = sum_K Wt[M,K] * X[K,N] + bias, K=64.
// A-fragment (Wt, ISA 16x4 f32 layout): lane L holds
//   A[L%16, k0 + 2*(L/16) + {0,1}] = W_m[k0+2*(L/16)+{0,1}, 16m + L%16]
// B-fragment (X, 4x16): lane L holds X[k0 + 2*(L/16) + {x:0,y:1}, p0 + L%16]
//   -> per-component load is 16 consecutive floats per half-wave (coalesced).
// C/D (16x16 f32): lane L, VGPR v holds D[v + 8*(L/16), L%16].
__global__ __launch_bounds__(256, 1)
void astar_pixel_linear_wmma(const float* __restrict__ feat,
                             const float* __restrict__ Wm,
                             const float* __restrict__ bm,
                             const unsigned char* __restrict__ flags,
                             float* __restrict__ out) {
    const int lane  = threadIdx.x & 31;
    const int half  = (lane >> 4) & 1;     // 0: lanes 0-15, 1: lanes 16-31
    const int nidx  = lane & 15;
    const int wave  = ((blockIdx.x % BLOCKS_PER_BATCH) << 3) | (threadIdx.x >> 5);
    const int batch = blockIdx.x / BLOCKS_PER_BATCH;

    // --- W^T fragments: resident in VGPRs for the whole kernel (128 VGPRs).
    v2f wfrag[4][16];
#pragma unroll
    for (int m = 0; m < 4; ++m) {
#pragma unroll
        for (int kk = 0; kk < 16; ++kk) {
            const int k   = kk * 4 + half * 2;
            const int col = m * 16 + nidx;
            v2f w;
            w.x = Wm[k * CCH + col];
            w.y = Wm[(k + 1) * CCH + col];
            wfrag[m][kk] = w;
        }
    }

    // --- bias fragments: acc-layout element (lane L, v) is channel v+8*half.
    v8f bias[4];
#pragma unroll
    for (int m = 0; m < 4; ++m)
        bias[m] = *(const v8f*)(bm + m * 16 + half * 8);

    const float*         fb  = feat  + (size_t)batch * CCH * HWN;
    float*               ob  = out   + (size_t)batch * CCH * HWN;
    const unsigned char* flb = flags + (size_t)batch * HWN;

    // 2304 tiles / 64 waves = 36 tiles each: uniform trip count, EXEC all-1s.
    for (int t = wave; t < TILES_PER_BATCH; t += WAVES_PER_BATCH) {
        const int p = (t << 4) + nidx;    // this lane's pixel (N index)

        // X tile: 64(K) x 16(pixels), 32 coalesced b32 loads per lane.
        v2f xfrag[16];
#pragma unroll
        for (int kk = 0; kk < 16; ++kk) {
            const int k = kk * 4 + half * 2;
            v2f x;
            x.x = fb[(size_t)k * HWN + p];
            x.y = fb[(size_t)(k + 1) * HWN + p];
            xfrag[kk] = x;
        }

        const float fmask = flb[p] ? 1.0f : 0.0f;   // visited-pixel mask

#pragma unroll
        for (int m = 0; m < 4; ++m) {               // 4 output-channel tiles
            v8f acc = bias[m];
#pragma unroll
            for (int kk = 0; kk < 16; ++kk) {       // K = 64 in steps of 4
                acc = __builtin_amdgcn_wmma_f32_16x16x4_f32(
                    /*neg_a=*/false, wfrag[m][kk],
                    /*neg_b=*/false, xfrag[kk],
                    /*c_mod=*/(short)0, acc,
                    /*reuse_a=*/false, /*reuse_b=*/false);
            }
#pragma unroll
            for (int v = 0; v < 8; ++v) {
                // out[b, 16m + v + 8*half, p]; 64B contiguous per half-wave
                ob[(size_t)(m * 16 + v + half * 8) * HWN + p] = acc[v] * fmask;
            }
        }
    }
}

// ---------------------------------------------------------------------------
extern "C" void kernel_launch(void* const* d_in, const int* in_sizes, int n_in,
                              void* d_out, int out_size, void* d_ws, size_t ws_size,
                              hipStream_t stream) {
    const float*         feat     = (const float*)d_in[0];
    const int*           path_idx = (const int*)d_in[1];
    const float*         Wm       = (const float*)d_in[2];
    const float*         bm       = (const float*)d_in[3];
    float*               out      = (float*)d_out;
    unsigned char*       flags    = (unsigned char*)d_ws;  // B*HW = 589,824 B

    // 1) clear visit flags (uint4 stores, exact grid: 36864 threads)
    const int n4 = (BATCHES * HWN) / 16;
    astar_zero_flags<<<(n4 + 255) / 256, 256, 0, stream>>>((uint4*)d_ws, n4);

    // 2) mark visited pixels (B*P*L = 1,048,576 threads, exact grid)
    astar_set_flags<<<(BATCHES * PLN) / 256, 256, 0, stream>>>(path_idx, flags);

    // 3) masked 64x64 linear map over all pixels via f32 WMMA
    astar_pixel_linear_wmma<<<BATCHES * BLOCKS_PER_BATCH, 256, 0, stream>>>(
        feat, Wm, bm, flags, out);
}